// CrossScaleVFE_13984413516425
// MI455X (gfx1250) — compile-verified
//
#include <hip/hip_runtime.h>

// CrossScaleVFE on MI455X (gfx1250, wave32).
// Batched 16x16 fp32 GEMMs via V_WMMA_F32_16X16X4_F32, per-pair Cholesky/KL in LDS.

typedef __attribute__((ext_vector_type(2))) float v2f;
typedef __attribute__((ext_vector_type(8))) float v8f;

#define KDIM   16
#define NCHILD 2048
#define MPAR   64

// ---------------------------------------------------------------------------
// WMMA helpers: 16x16 fp32 matmul = 4 chained V_WMMA_F32_16X16X4_F32.
// A-frag (16x4, MxK): lane l: m=l&15, half=l>>4; a[v] = X[m][kk + 2*half + v]
// B-frag (4x16, KxN): lane l: n=l&15;            b[v] = Y[kk + 2*half + v][n]
// C/D   (16x16):      vgpr r holds D[r + 8*half][n]
// ---------------------------------------------------------------------------

__device__ __forceinline__ v8f wmma16_AB(const float* __restrict__ X,
                                         const float* __restrict__ Y) {
  const int lane = threadIdx.x & 31;
  const int half = lane >> 4;
  const int r    = lane & 15;
  v8f c = {};
#pragma unroll
  for (int kk = 0; kk < 16; kk += 4) {
    const int ka = kk + 2 * half;
    v2f a, b;
    a[0] = X[r * 16 + ka];
    a[1] = X[r * 16 + ka + 1];
    b[0] = Y[ka * 16 + r];
    b[1] = Y[(ka + 1) * 16 + r];
    c = __builtin_amdgcn_wmma_f32_16x16x4_f32(false, a, false, b,
                                              (short)0, c, false, false);
  }
  return c;
}

// C = X * T^T : B-frag of T^T is the A-frag addressing pattern applied to T.
__device__ __forceinline__ v8f wmma16_ABt(const float* __restrict__ X,
                                          const float* __restrict__ T) {
  const int lane = threadIdx.x & 31;
  const int half = lane >> 4;
  const int r    = lane & 15;
  v8f c = {};
#pragma unroll
  for (int kk = 0; kk < 16; kk += 4) {
    const int ka = kk + 2 * half;
    v2f a, b;
    a[0] = X[r * 16 + ka];
    a[1] = X[r * 16 + ka + 1];
    b[0] = T[r * 16 + ka];      // T^T[ka][n] = T[n][ka], n == r
    b[1] = T[r * 16 + ka + 1];
    c = __builtin_amdgcn_wmma_f32_16x16x4_f32(false, a, false, b,
                                              (short)0, c, false, false);
  }
  return c;
}

__device__ __forceinline__ void store_frag(float* __restrict__ D, v8f c) {
  const int lane = threadIdx.x & 31;
  const int half = lane >> 4;
  const int r    = lane & 15;
#pragma unroll
  for (int rr = 0; rr < 8; ++rr)
    D[(rr + 8 * half) * 16 + r] = c[rr];
}

// ---------------------------------------------------------------------------
// Wave-cooperative in-place Cholesky of a 16x16 SPD matrix in LDS (lower tri).
// Block must be a single wave of 32 threads.
// ---------------------------------------------------------------------------
__device__ __forceinline__ void cholesky16(float* S) {
  const int lane = threadIdx.x;
  for (int k = 0; k < 16; ++k) {
    __syncthreads();
    if (lane == 0) S[k * 16 + k] = sqrtf(S[k * 16 + k]);
    __syncthreads();
    const float dk = S[k * 16 + k];
    if (lane > k && lane < 16) S[lane * 16 + k] /= dk;
    __syncthreads();
    if (lane > k && lane < 16) {
      const float lik = S[lane * 16 + k];
      for (int j = k + 1; j <= lane; ++j)
        S[lane * 16 + j] -= lik * S[j * 16 + k];
    }
  }
  __syncthreads();
}

// ---------------------------------------------------------------------------
// Kernel 1: invert 16x16 gauge frames (Gauss-Jordan, no pivoting: Omega = I + eps).
// One wave per matrix; 32 lanes <-> 32 augmented columns.
// ---------------------------------------------------------------------------
__global__ void __launch_bounds__(32)
invert_gauge_kernel(const float* __restrict__ omega, float* __restrict__ oinv) {
  const int m    = blockIdx.x;
  const int lane = threadIdx.x;
  __shared__ float A[16 * 32];
  const float* src = omega + m * 256;
#pragma unroll
  for (int i = 0; i < 16; ++i) {
    float v;
    if (lane < 16) v = src[i * 16 + lane];
    else           v = (lane - 16 == i) ? 1.0f : 0.0f;
    A[i * 32 + lane] = v;
  }
  __syncthreads();
  for (int k = 0; k < 16; ++k) {
    const float piv = A[k * 32 + k];
    __syncthreads();
    A[k * 32 + lane] /= piv;               // scale pivot row (own column)
    __syncthreads();
    float f[16];
#pragma unroll
    for (int i = 0; i < 16; ++i) f[i] = A[i * 32 + k];   // factors (col k)
    __syncthreads();
#pragma unroll
    for (int i = 0; i < 16; ++i)
      if (i != k) A[i * 32 + lane] -= f[i] * A[k * 32 + lane];
    __syncthreads();
  }
  if (lane >= 16) {
#pragma unroll
    for (int i = 0; i < 16; ++i)
      oinv[m * 256 + i * 16 + (lane - 16)] = A[i * 32 + lane];
  }
}

// ---------------------------------------------------------------------------
// Kernel 2: logdet of child covariances via Cholesky. One wave per matrix.
// ---------------------------------------------------------------------------
__global__ void __launch_bounds__(32)
child_logdet_kernel(const float* __restrict__ sigma, float* __restrict__ logdet) {
  const int n    = blockIdx.x;
  const int lane = threadIdx.x;
  __shared__ float S[256];
  const float* src = sigma + n * 256;
  for (int idx = lane; idx < 256; idx += 32) S[idx] = src[idx];
  __syncthreads();
  cholesky16(S);
  if (lane == 0) {
    float ld = 0.0f;
#pragma unroll
    for (int k = 0; k < 16; ++k) ld += __logf(S[k * 16 + k]);
    logdet[n] = 2.0f * ld;
  }
}

// ---------------------------------------------------------------------------
// Kernel 3: per-(n,m) weighted KL. One wave per pair.
//   T  = Omega_c[n] * OmegaP_inv[m]          (WMMA x4)
//   U  = T * Sigma_p[m]                      (WMMA x4)
//   St = U * T^T                             (WMMA x4)
//   KL = 0.5*( tr(St^-1 Sc) + d^T St^-1 d - 16 + logdet St - logdet Sc )
// ---------------------------------------------------------------------------
__global__ void __launch_bounds__(32)
pair_kl_kernel(const float* __restrict__ mu_c,  const float* __restrict__ sig_c,
               const float* __restrict__ omg_c,
               const float* __restrict__ mu_par, const float* __restrict__ sig_par,
               const float* __restrict__ opinv,  const float* __restrict__ logdet_c,
               const float* __restrict__ W,      float* __restrict__ accum) {
  const int m    = blockIdx.x;   // 0..63
  const int n    = blockIdx.y;   // 0..2047
  const int lane = threadIdx.x;

  __shared__ float T[256];
  __shared__ float U[256];
  __shared__ float St[256];
  __shared__ float diffS[16];
  __shared__ float red[32];

  // --- three fp32 WMMA GEMMs (no divergence before these; EXEC all ones) ---
  v8f t = wmma16_AB(omg_c + n * 256, opinv + m * 256);
  store_frag(T, t);
  __syncthreads();
  v8f u = wmma16_AB(T, sig_par + m * 256);
  store_frag(U, u);
  __syncthreads();
  v8f s = wmma16_ABt(U, T);
  store_frag(St, s);

  // mu_t = T * mu_parent ; diff = mu_t - mu_child
  if (lane < 16) {
    float acc = 0.0f;
#pragma unroll
    for (int j = 0; j < 16; ++j) acc += T[lane * 16 + j] * mu_par[m * 16 + j];
    diffS[lane] = acc - mu_c[n * 16 + lane];
  }
  __syncthreads();

  cholesky16(St);   // St -> L (lower triangle), includes barriers

  // 17 independent triangular solves: lanes 0..15 -> columns of Sigma_c
  // (trace term), lane 16 -> diff (Mahalanobis term).
  float contrib = 0.0f;
  if (lane < 17) {
    float y[16];
#pragma unroll
    for (int i = 0; i < 16; ++i)
      y[i] = (lane < 16) ? sig_c[n * 256 + i * 16 + lane] : diffS[i];
    // forward: L y = b
#pragma unroll
    for (int i = 0; i < 16; ++i) {
      float a = y[i];
#pragma unroll
      for (int j = 0; j < i; ++j) a -= St[i * 16 + j] * y[j];
      y[i] = a / St[i * 16 + i];
    }
    // backward: L^T x = y
#pragma unroll
    for (int i = 15; i >= 0; --i) {
      float a = y[i];
#pragma unroll
      for (int j = i + 1; j < 16; ++j) a -= St[j * 16 + i] * y[j];
      y[i] = a / St[i * 16 + i];
    }
    if (lane < 16) {
      contrib = y[lane];              // diagonal entry of St^-1 * Sc
    } else {
      float mh = 0.0f;
#pragma unroll
      for (int i = 0; i < 16; ++i) mh += diffS[i] * y[i];
      contrib = mh;
    }
  }
  red[lane] = contrib;
  __syncthreads();

  if (lane == 0) {
    float tr = 0.0f;
#pragma unroll
    for (int c = 0; c < 16; ++c) tr += red[c];
    const float maha = red[16];
    float ldt = 0.0f;
#pragma unroll
    for (int k = 0; k < 16; ++k) ldt += __logf(St[k * 16 + k]);
    ldt *= 2.0f;
    const float kl = 0.5f * (tr + maha - 16.0f + ldt - logdet_c[n]);
    atomicAdd(accum, W[n * MPAR + m] * kl);
  }
}

__global__ void init_out_kernel(float* out) {
  if (threadIdx.x == 0) { out[0] = 0.0f; out[1] = 0.0f; out[2] = 0.0f; }
}
__global__ void finalize_kernel(float* out) {
  if (threadIdx.x == 0) out[0] = out[1] + out[2];  // lambdas are 1.0
}

// ---------------------------------------------------------------------------
extern "C" void kernel_launch(void* const* d_in, const int* in_sizes, int n_in,
                              void* d_out, int out_size, void* d_ws, size_t ws_size,
                              hipStream_t stream) {
  (void)in_sizes; (void)n_in; (void)out_size; (void)ws_size;
  const float* mu_p   = (const float*)d_in[0];
  const float* sig_p  = (const float*)d_in[1];
  const float* mu_r   = (const float*)d_in[2];
  const float* sig_r  = (const float*)d_in[3];
  const float* omg_c  = (const float*)d_in[4];
  const float* omg_mc = (const float*)d_in[5];
  const float* mu_q   = (const float*)d_in[6];
  const float* sig_q  = (const float*)d_in[7];
  const float* mu_s   = (const float*)d_in[8];
  const float* sig_s  = (const float*)d_in[9];
  const float* omg_p  = (const float*)d_in[10];
  const float* omg_mp = (const float*)d_in[11];
  const float* W      = (const float*)d_in[12];
  float* out = (float*)d_out;

  float* ws     = (float*)d_ws;
  float* opinv0 = ws;                      // 64*256
  float* opinv1 = ws + MPAR * 256;         // 64*256
  float* ldc0   = ws + 2 * MPAR * 256;     // 2048
  float* ldc1   = ldc0 + NCHILD;           // 2048  (total ~144 KB)

  init_out_kernel<<<1, 32, 0, stream>>>(out);

  invert_gauge_kernel<<<MPAR, 32, 0, stream>>>(omg_p,  opinv0);
  invert_gauge_kernel<<<MPAR, 32, 0, stream>>>(omg_mp, opinv1);
  child_logdet_kernel<<<NCHILD, 32, 0, stream>>>(sig_p, ldc0);
  child_logdet_kernel<<<NCHILD, 32, 0, stream>>>(sig_r, ldc1);

  dim3 grid(MPAR, NCHILD);
  pair_kl_kernel<<<grid, 32, 0, stream>>>(mu_p, sig_p, omg_c,  mu_q, sig_q,
                                          opinv0, ldc0, W, out + 1);
  pair_kl_kernel<<<grid, 32, 0, stream>>>(mu_r, sig_r, omg_mc, mu_s, sig_s,
                                          opinv1, ldc1, W, out + 2);

  finalize_kernel<<<1, 32, 0, stream>>>(out);
}